// QuickShiftGroups_35802847379477
// MI455X (gfx1250) — compile-verified
//
#include <hip/hip_runtime.h>
#include <math.h>

typedef __attribute__((ext_vector_type(2))) float v2f;
typedef __attribute__((ext_vector_type(8))) float v8f;

#define H_    224
#define W_    224
#define NIMG  4
#define PLANE (H_ * W_)
#define TPR   14      // 224/16 center tiles per row
#define NJUMP 17      // ceil(log2(50176)) + 1

// -0.5/KERNEL_SIZE^2 * log2(e): exp(-0.02*d) == exp2(C2_*d)
#define C2_   (-0.028853900817779268f)
#define K3_   (0.057707801635558536f)   // -2 * C2_

__device__ __forceinline__ int reflect_idx(int i, int L) {
  if (i < 0) i = -i - 1;
  if (i >= L) i = 2 * L - 1 - i;
  return i;
}

// Separable Gaussian smooth, one pass (axisH=1: along H, else along W).
__global__ void k_smooth(const float* __restrict__ in, float* __restrict__ out, int axisH) {
  int idx = blockIdx.x * blockDim.x + threadIdx.x;
  const int total = NIMG * 3 * PLANE;
  if (idx >= total) return;
  int w = idx % W_;
  int h = (idx / W_) % H_;
  int pc = idx / PLANE;           // plane index n*3+c
  // weights: radius 12, sigma 3, normalized (computed in f64 like numpy)
  double kd[25];
  double s = 0.0;
  #pragma unroll
  for (int t = 0; t < 25; ++t) {
    double u = (double)(t - 12) / 3.0;
    kd[t] = exp(-0.5 * u * u);
    s += kd[t];
  }
  const float* plane = in + (size_t)pc * PLANE;
  float acc = 0.0f;
  #pragma unroll
  for (int t = 0; t < 25; ++t) {
    float kf = (float)(kd[t] / s);
    int hh = h, ww = w;
    if (axisH) hh = reflect_idx(h + t - 12, H_);
    else       ww = reflect_idx(w + t - 12, W_);
    acc += kf * plane[hh * W_ + ww];
  }
  out[idx] = acc;
}

// Per-pixel squared norm over 3 channels.
__global__ void k_sqn(const float* __restrict__ sm, float* __restrict__ sqn) {
  int idx = blockIdx.x * blockDim.x + threadIdx.x;
  if (idx >= NIMG * PLANE) return;
  int n = idx / PLANE;
  int p = idx % PLANE;
  const float* b = sm + (size_t)n * 3 * PLANE;
  float a0 = b[p], a1 = b[PLANE + p], a2 = b[2 * PLANE + p];
  sqn[idx] = a0 * a0 + a1 * a1 + a2 * a2;
}

// Density via exact-f32 WMMA gram tiles. One wave == 16 centers in a row.
// Row validity handled by scalar loop bounds; column validity by -INF constants.
__global__ __launch_bounds__(256) void k_density(const float* __restrict__ sm,
                                                 const float* __restrict__ sqn,
                                                 float* __restrict__ dens) {
  int wid  = __builtin_amdgcn_readfirstlane(
                 blockIdx.x * (blockDim.x >> 5) + (threadIdx.x >> 5));
  int lane = threadIdx.x & 31;
  int nimg = wid / (TPR * H_);
  int rem  = wid % (TPR * H_);
  int r    = rem / TPR;          // scalar
  int c0   = (rem % TPR) * 16;   // scalar
  int half = lane >> 4;          // 0: K={0,1} (ch0,ch1); 1: K={2,3} (ch2,0)
  int nn   = lane & 15;

  const float* p0 = sm + (size_t)(nimg * 3 + 0) * PLANE;
  const float* p1 = sm + (size_t)(nimg * 3 + 1) * PLANE;
  const float* p2 = sm + (size_t)(nimg * 3 + 2) * PLANE;
  const float* sq = sqn + (size_t)nimg * PLANE;
  const float* pX = half ? p2 : p0;   // per-lane base for first K component
  const float  FINF = __builtin_inff();

  // A-matrix: 16 centers x 4 channels (K=3 padded with zero)
  int mc = c0 + nn;
  v2f A;
  {
    float a0 = pX[r * W_ + mc];
    float a1 = p1[r * W_ + mc];
    A.x = a0;
    A.y = half ? 0.0f : a1;
  }

  // dr-invariant per-(g,v) geometry, pre-scaled by C2_; invalid -> -INF
  int   colc_[3];
  float q2_[3][8];
  #pragma unroll
  for (int g = 0; g < 3; ++g) {
    int j = g * 16 + nn;
    int col = c0 - 15 + j;
    bool colok = (col >= 0) && (col < W_);
    colc_[g] = col < 0 ? 0 : (col >= W_ ? W_ - 1 : col);
    #pragma unroll
    for (int v = 0; v < 8; ++v) {
      int m = v + (half << 3);
      int dc = j - m - 15;
      bool ok = colok && (dc >= -15) && (dc <= 15);
      q2_[g][v] = ok ? (C2_ * (float)(dc * dc)) : -FINF;
    }
  }

  float acc[8];
  #pragma unroll
  for (int v = 0; v < 8; ++v) acc[v] = 0.0f;

  int dr0 = (r >= 15) ? -15 : -r;
  int dr1 = (r + 15 < H_) ? 15 : (H_ - 1 - r);
  for (int dr = dr0; dr <= dr1; ++dr) {
    int rr = r + dr;               // always in range
    float drsq2 = C2_ * (float)(dr * dr);
    #pragma unroll
    for (int g = 0; g < 3; ++g) {
      int qi = rr * W_ + colc_[g];
      float b0 = pX[qi];
      float b1 = p1[qi];
      float sqq = sq[qi];
      v2f B;
      B.x = b0;
      B.y = half ? 0.0f : b1;
      v8f C = {};
      v8f D = __builtin_amdgcn_wmma_f32_16x16x4_f32(false, A, false, B,
                                                    (short)0, C, false, false);
      float base = drsq2 + C2_ * sqq;
      #pragma unroll
      for (int v = 0; v < 8; ++v) {
        float t = base + q2_[g][v];                       // -INF if invalid
        acc[v] += __builtin_amdgcn_exp2f(__builtin_fmaf(K3_, D[v], t));
      }
    }
  }
  // reduce across the 16 lanes of each half-wave
  #pragma unroll
  for (int v = 0; v < 8; ++v) {
    float a = acc[v];
    a += __shfl_xor(a, 1, 32);
    a += __shfl_xor(a, 2, 32);
    a += __shfl_xor(a, 4, 32);
    a += __shfl_xor(a, 8, 32);
    acc[v] = a;
  }
  if (nn < 8) {
    int m = nn + (half << 3);
    int cm = c0 + m;
    float outv = 0.0f;
    #pragma unroll
    for (int v = 0; v < 8; ++v) outv = (nn == v) ? acc[v] : outv;
    float sp = sq[r * W_ + cm];
    dens[(size_t)nimg * PLANE + r * W_ + cm] = outv * __builtin_amdgcn_exp2f(C2_ * sp);
  }
}

// Parent selection: nearest higher-density neighbor (k-index tie-break).
__global__ __launch_bounds__(256) void k_parent(const float* __restrict__ sm,
                                                const float* __restrict__ sqn,
                                                const float* __restrict__ dens,
                                                int* __restrict__ parent) {
  int wid  = __builtin_amdgcn_readfirstlane(
                 blockIdx.x * (blockDim.x >> 5) + (threadIdx.x >> 5));
  int lane = threadIdx.x & 31;
  int nimg = wid / (TPR * H_);
  int rem  = wid % (TPR * H_);
  int r    = rem / TPR;
  int c0   = (rem % TPR) * 16;
  int half = lane >> 4;
  int nn   = lane & 15;

  const float* p0 = sm + (size_t)(nimg * 3 + 0) * PLANE;
  const float* p1 = sm + (size_t)(nimg * 3 + 1) * PLANE;
  const float* p2 = sm + (size_t)(nimg * 3 + 2) * PLANE;
  const float* sq = sqn  + (size_t)nimg * PLANE;
  const float* dn = dens + (size_t)nimg * PLANE;
  const float* pX = half ? p2 : p0;
  const float  FINF = __builtin_inff();

  int mc = c0 + nn;
  v2f A;
  {
    float a0 = pX[r * W_ + mc];
    float a1 = p1[r * W_ + mc];
    A.x = a0;
    A.y = half ? 0.0f : a1;
  }

  float densC[8];
  #pragma unroll
  for (int v = 0; v < 8; ++v) densC[v] = dn[r * W_ + c0 + v + (half << 3)];

  // dr-invariant per-(g,v): dc^2 + |center|^2, +INF when geometry invalid
  int   colc_[3];
  float qs_[3][8];
  int   kc_[3][8];      // dc + 15 (column part of k index)
  #pragma unroll
  for (int g = 0; g < 3; ++g) {
    int j = g * 16 + nn;
    int col = c0 - 15 + j;
    bool colok = (col >= 0) && (col < W_);
    colc_[g] = col < 0 ? 0 : (col >= W_ ? W_ - 1 : col);
    #pragma unroll
    for (int v = 0; v < 8; ++v) {
      int m = v + (half << 3);
      int dc = j - m - 15;
      bool ok = colok && (dc >= -15) && (dc <= 15);
      float sqc = sq[r * W_ + c0 + m];
      qs_[g][v] = ok ? ((float)(dc * dc) + sqc) : FINF;
      kc_[g][v] = dc + 15;
    }
  }

  float best[8];
  int bestk[8];
  #pragma unroll
  for (int v = 0; v < 8; ++v) { best[v] = FINF; bestk[v] = 0x7fffffff; }

  int dr0 = (r >= 15) ? -15 : -r;
  int dr1 = (r + 15 < H_) ? 15 : (H_ - 1 - r);
  for (int dr = dr0; dr <= dr1; ++dr) {
    int rr = r + dr;
    float drsq = (float)(dr * dr);
    int krow = (dr + 15) * 31;
    #pragma unroll
    for (int g = 0; g < 3; ++g) {
      int qi = rr * W_ + colc_[g];
      float b0 = pX[qi];
      float b1 = p1[qi];
      float sqq = sq[qi];
      float dq  = dn[qi];
      v2f B;
      B.x = b0;
      B.y = half ? 0.0f : b1;
      v8f C = {};
      v8f D = __builtin_amdgcn_wmma_f32_16x16x4_f32(false, A, false, B,
                                                    (short)0, C, false, false);
      float base = drsq + sqq;
      #pragma unroll
      for (int v = 0; v < 8; ++v) {
        float t = base + qs_[g][v];                        // +INF if invalid
        float dist = __builtin_fmaf(-2.0f, D[v], t);       // INF stays INF
        float cd = (dq > densC[v]) ? dist : FINF;          // per-v cndmask
        // per-lane visit order has strictly increasing k -> strict < keeps first k
        bool better = cd < best[v];
        best[v]  = better ? cd : best[v];
        bestk[v] = better ? (krow + kc_[g][v]) : bestk[v];
      }
    }
  }
  // argmin reduction across each 16-lane half (min dist, then min k)
  #pragma unroll
  for (int v = 0; v < 8; ++v) {
    float b = best[v];
    int k = bestk[v];
    #pragma unroll
    for (int msk = 1; msk < 16; msk <<= 1) {
      float ob = __shfl_xor(b, msk, 32);
      int ok_ = __shfl_xor(k, msk, 32);
      bool take = (ob < b) || ((ob == b) && (ok_ < k));
      b = take ? ob : b;
      k = take ? ok_ : k;
    }
    best[v] = b;
    bestk[v] = k;
  }
  if (nn < 8) {
    int m = nn + (half << 3);
    int cm = c0 + m;
    float b = FINF;
    int k = 0;
    #pragma unroll
    for (int v = 0; v < 8; ++v) {
      b = (nn == v) ? best[v] : b;
      k = (nn == v) ? bestk[v] : k;
    }
    int self = r * W_ + cm;
    int ddr = k / 31 - 15;
    int ddc = k % 31 - 15;
    int cand = (r + ddr) * W_ + (cm + ddc);
    int par = (b <= 100.0f) ? cand : self;   // sqrt(dist) <= MAX_DIST
    parent[(size_t)nimg * PLANE + self] = par;
  }
}

// One pointer-jumping step: p_out[i] = p_in[p_in[i]] (per-image flat indices).
__global__ void k_jump(const int* __restrict__ pin, int* __restrict__ pout) {
  int idx = blockIdx.x * blockDim.x + threadIdx.x;
  if (idx >= NIMG * PLANE) return;
  int n = idx / PLANE;
  int p = pin[idx];
  pout[idx] = pin[n * PLANE + p];
}

// Root labeling: order = cumsum(roots)-1 (blocked scan), labels = min(order[parent],39).
// One workgroup per image; 50176 = 1024 * 49 exactly.
__global__ __launch_bounds__(1024) void k_label(const int* __restrict__ parent,
                                                int* __restrict__ order,
                                                float* __restrict__ out) {
  __shared__ int part[1024];
  const int CH = 49;
  int n = blockIdx.x;
  int t = threadIdx.x;
  int base = t * CH;
  const int* par = parent + (size_t)n * PLANE;
  int cnt = 0;
  for (int j = 0; j < CH; ++j) cnt += (par[base + j] == base + j) ? 1 : 0;
  part[t] = cnt;
  __syncthreads();
  for (int off = 1; off < 1024; off <<= 1) {
    int add = (t >= off) ? part[t - off] : 0;
    __syncthreads();
    part[t] += add;
    __syncthreads();
  }
  int run = part[t] - cnt;  // exclusive prefix of this chunk
  int* ordn = order + (size_t)n * PLANE;
  for (int j = 0; j < CH; ++j) {
    run += (par[base + j] == base + j) ? 1 : 0;
    ordn[base + j] = run - 1;
  }
  __threadfence_block();
  __syncthreads();
  float* on = out + (size_t)n * PLANE;
  for (int j = 0; j < CH; ++j) {
    int lab = ordn[par[base + j]];
    if (lab > 39) lab = 39;
    on[base + j] = (float)lab;
  }
}

extern "C" void kernel_launch(void* const* d_in, const int* in_sizes, int n_in,
                              void* d_out, int out_size, void* d_ws, size_t ws_size,
                              hipStream_t stream) {
  const float* x = (const float*)d_in[0];
  float* out = (float*)d_out;

  float* ws   = (float*)d_ws;
  float* tmp  = ws;                          // N*3*H*W
  float* sm   = tmp + (size_t)NIMG * 3 * PLANE;
  float* sqn  = sm  + (size_t)NIMG * 3 * PLANE;
  float* dens = sqn + (size_t)NIMG * PLANE;
  int* parA   = (int*)(dens + (size_t)NIMG * PLANE);
  int* parB   = parA + (size_t)NIMG * PLANE;
  int* order  = parB + (size_t)NIMG * PLANE;

  const int totalNC = NIMG * 3 * PLANE;
  const int totalP  = NIMG * PLANE;
  const int tiles   = NIMG * H_ * TPR;       // 12544, 8 waves/block

  k_smooth<<<(totalNC + 255) / 256, 256, 0, stream>>>(x, tmp, 1);
  k_smooth<<<(totalNC + 255) / 256, 256, 0, stream>>>(tmp, sm, 0);
  k_sqn<<<(totalP + 255) / 256, 256, 0, stream>>>(sm, sqn);
  k_density<<<tiles / 8, 256, 0, stream>>>(sm, sqn, dens);
  k_parent<<<tiles / 8, 256, 0, stream>>>(sm, sqn, dens, parA);

  int* pa = parA;
  int* pb = parB;
  for (int it = 0; it < NJUMP; ++it) {
    k_jump<<<(totalP + 255) / 256, 256, 0, stream>>>(pa, pb);
    int* sw = pa; pa = pb; pb = sw;
  }
  k_label<<<NIMG, 1024, 0, stream>>>(pa, order, out);
}